// checkerboard_13924283973959
// MI455X (gfx1250) — compile-verified
//
#include <hip/hip_runtime.h>
#include <stdint.h>

typedef double2 cplx;
typedef float v2f __attribute__((ext_vector_type(2)));
typedef float v8f __attribute__((ext_vector_type(8)));

// ---------------------------------------------------------------------------
// CDNA5 async global<->LDS helpers (gfx1250 ASYNCcnt path)
// ---------------------------------------------------------------------------
__device__ __forceinline__ unsigned lds_byte_addr(const void* p) {
  return (unsigned)(uintptr_t)p;  // low 32 bits of generic shared ptr = LDS offset
}

__device__ __forceinline__ void async_g2l(unsigned lds_off, const cplx* gaddr) {
  unsigned long long ga = (unsigned long long)(uintptr_t)gaddr;
  asm volatile("global_load_async_to_lds_b128 %0, %1, off"
               :: "v"(lds_off), "v"(ga) : "memory");
}

__device__ __forceinline__ void async_l2g(cplx* gaddr, unsigned lds_off) {
  unsigned long long ga = (unsigned long long)(uintptr_t)gaddr;
  asm volatile("global_store_async_from_lds_b128 %0, %1, off"
               :: "v"(ga), "v"(lds_off) : "memory");
}

__device__ __forceinline__ void wait_async0() {
  asm volatile("s_wait_asynccnt 0x0" ::: "memory");
}

// padded LDS index (breaks power-of-2 bank conflict patterns; 16B elements)
__device__ __forceinline__ int padidx(int t) { return t + (t >> 4); }

// ---------------------------------------------------------------------------
// Tile-bit helpers: a "group" of 16 amplitudes spans tile bits {P0,P1,P2,P3};
// the group id u fills the remaining tile bits in ascending order.
// ---------------------------------------------------------------------------
template<int NB, int P0, int P1, int P2, int P3>
__device__ __forceinline__ int scatter_base(int u) {
  int t = 0, ub = 0;
#pragma unroll
  for (int pos = 0; pos < NB; ++pos) {
    if (pos == P0 || pos == P1 || pos == P2 || pos == P3) continue;
    t |= ((u >> ub) & 1) << pos;
    ++ub;
  }
  return t;
}

template<int P0, int P1, int P2, int P3>
__device__ __forceinline__ int colbits(int c) {
  return ((c & 1) << P0) | (((c >> 1) & 1) << P1) |
         (((c >> 2) & 1) << P2) | (((c >> 3) & 1) << P3);
}

// ---------------------------------------------------------------------------
// WMMA round: apply composed round operator W = I + E to every group of the
// LDS tile.  Amplitudes stay fp64; the correction Y = X * E^T is computed in
// f32 with v_wmma_f32_16x16x4_f32 (E ~ 1e-4, so f32 error ~1e-11 of |x|).
// Per wave iteration: 16 groups (=256 amps), 16 WMMAs (4 real GEMMs, K=16).
// ---------------------------------------------------------------------------
template<int NB, int P0, int P1, int P2, int P3>
__device__ __forceinline__ void wmma_round(cplx* tile, int lane, int wave,
                                           int nIter, const float* __restrict__ E) {
  const float* Er  = E;         // Re(E), row-major [out n][in c]
  const float* Ei  = E + 256;   // Im(E)
  const float* Eni = E + 512;   // -Im(E)
  const int n    = lane & 15;          // matrix row (A) / column (B,D)
  const int off0 = (lane >> 4) << 1;   // K offset for this half-wave

  // B operands (E^T chunks), loop-invariant: B[k][n] = E[n][k]
  v2f ber[4], bei[4], bnei[4];
#pragma unroll
  for (int c = 0; c < 4; ++c) {
    const int k0 = 4 * c + off0;
    ber[c].x  = Er [n * 16 + k0]; ber[c].y  = Er [n * 16 + k0 + 1];
    bei[c].x  = Ei [n * 16 + k0]; bei[c].y  = Ei [n * 16 + k0 + 1];
    bnei[c].x = Eni[n * 16 + k0]; bnei[c].y = Eni[n * 16 + k0 + 1];
  }

  for (int it = 0; it < nIter; ++it) {
    const int gset = (wave * nIter + it) << 4;  // 16 groups per iteration

    // --- gather A operands: lane covers row m=n, its half-wave K columns ---
    const int baseA = scatter_base<NB, P0, P1, P2, P3>(gset + n);
    v2f xr[4], xi[4];
#pragma unroll
    for (int c = 0; c < 4; ++c) {
      const int c0 = 4 * c + off0;
      const cplx e0 = tile[padidx(baseA | colbits<P0, P1, P2, P3>(c0))];
      const cplx e1 = tile[padidx(baseA | colbits<P0, P1, P2, P3>(c0 + 1))];
      xr[c].x = (float)e0.x; xr[c].y = (float)e1.x;
      xi[c].x = (float)e0.y; xi[c].y = (float)e1.y;
    }

    // --- 4 real GEMMs: Yr = Xr*Er^T - Xi*Ei^T ; Yi = Xr*Ei^T + Xi*Er^T ---
    v8f accr = {};
    v8f acci = {};
#pragma unroll
    for (int c = 0; c < 4; ++c)
      accr = __builtin_amdgcn_wmma_f32_16x16x4_f32(false, xr[c], false, ber[c],
                                                   (short)0, accr, false, false);
#pragma unroll
    for (int c = 0; c < 4; ++c)
      accr = __builtin_amdgcn_wmma_f32_16x16x4_f32(false, xi[c], false, bnei[c],
                                                   (short)0, accr, false, false);
#pragma unroll
    for (int c = 0; c < 4; ++c)
      acci = __builtin_amdgcn_wmma_f32_16x16x4_f32(false, xr[c], false, bei[c],
                                                   (short)0, acci, false, false);
#pragma unroll
    for (int c = 0; c < 4; ++c)
      acci = __builtin_amdgcn_wmma_f32_16x16x4_f32(false, xi[c], false, ber[c],
                                                   (short)0, acci, false, false);

    // --- scatter: x += correction (fp64 accumulate), D layout rows v/(v+8) ---
#pragma unroll
    for (int v = 0; v < 8; ++v) {
      const int row = v + ((lane >> 4) << 3);
      const int t = scatter_base<NB, P0, P1, P2, P3>(gset + row) |
                    colbits<P0, P1, P2, P3>(n);
      cplx o = tile[padidx(t)];
      o.x += (double)accr[v];
      o.y += (double)acci[v];
      tile[padidx(t)] = o;
    }
  }
}

// ---------------------------------------------------------------------------
// expm of 48 4x4 anti-Hermitian matrices (Taylor-Horner order 12, fp64).
// U layout in ws: [(layer*2 + offset)*12 + i] * 16 complex, offset0 = H1.
// ---------------------------------------------------------------------------
__global__ __launch_bounds__(64) void expm_kernel(const cplx* __restrict__ H1,
                                                  const cplx* __restrict__ H2,
                                                  cplx* __restrict__ U) {
  const int m = threadIdx.x;
  if (m >= 48) return;
  const int layer  = m / 24;
  const int offset = (m / 12) & 1;
  const int i      = m % 12;
  const cplx* H = (offset == 0 ? H1 : H2) + (layer * 12 + i) * 16;

  double ar[16], ai[16];
  for (int r = 0; r < 4; ++r)
    for (int c = 0; c < 4; ++c) {
      cplx h = H[r * 4 + c], ht = H[c * 4 + r];
      ar[r * 4 + c] = h.x - ht.x;
      ai[r * 4 + c] = h.y + ht.y;
    }

  double tr[16], ti[16];
  for (int e = 0; e < 16; ++e) { tr[e] = (e % 5 == 0) ? 1.0 : 0.0; ti[e] = 0.0; }

  for (int k = 12; k >= 1; --k) {
    double mr[16], mi[16];
    for (int r = 0; r < 4; ++r)
      for (int c = 0; c < 4; ++c) {
        double sr = 0.0, si = 0.0;
        for (int q = 0; q < 4; ++q) {
          double xr = ar[r * 4 + q], xi = ai[r * 4 + q];
          double br = tr[q * 4 + c], bi = ti[q * 4 + c];
          sr += xr * br - xi * bi;
          si += xr * bi + xi * br;
        }
        mr[r * 4 + c] = sr; mi[r * 4 + c] = si;
      }
    const double inv = 1.0 / (double)k;
    for (int e = 0; e < 16; ++e) {
      tr[e] = ((e % 5 == 0) ? 1.0 : 0.0) + mr[e] * inv;
      ti[e] = mi[e] * inv;
    }
  }
  cplx* Uo = U + m * 16;
  for (int e = 0; e < 16; ++e) Uo[e] = make_double2(tr[e], ti[e]);
}

// ---------------------------------------------------------------------------
// Compose each round's 16x16 operator W (fp64, exact), store E = W - I in f32
// (Re, Im, -Im), 768 floats per round-op.  11 round-ops per layer:
//   rid 0..3  : pass A rounds   (reg-bit gates per table below)
//   rid 4..6  : pass B rounds
//   rid 7..10 : pass B' rounds (rid 10 = wrap gate (q1=23,q2=0))
// ---------------------------------------------------------------------------
__global__ __launch_bounds__(32) void roundop_kernel(const cplx* __restrict__ Uws,
                                                     float* __restrict__ Eb) {
  const int ro = threadIdx.x;
  if (ro >= 22) return;
  const int layer = ro / 11;
  const int rid   = ro % 11;

  const int NG[11] = {3, 3, 3, 2, 2, 2, 2, 2, 2, 2, 1};
  // {bm, bl, offset(0=H1,1=H2), i}
  const int GT[11][3][4] = {
    {{1,0,0,11}, {3,2,0,10}, {2,1,1,10}},   // A R1: bits {0,1,2,3}
    {{1,0,0, 9}, {3,2,0, 8}, {2,1,1, 8}},   // A R2: bits {4,5,6,7}
    {{1,0,0, 7}, {3,2,0, 6}, {2,1,1, 6}},   // A R3: bits {8,9,10,11}
    {{1,0,1, 9}, {3,2,1, 7}, {0,0,0, 0}},   // A R4: bits {3,4,7,8}
    {{1,0,0, 5}, {3,2,0, 4}, {0,0,0, 0}},   // B R1: bits {7,8,9,10}
    {{3,2,0, 3}, {1,0,1, 5}, {0,0,0, 0}},   // B R2: bits {6,7,11,12}
    {{1,0,1, 4}, {3,2,1, 3}, {0,0,0, 0}},   // B R3: bits {8,9,10,11}
    {{1,0,0, 2}, {3,2,0, 1}, {0,0,0, 0}},   // B' R1
    {{3,2,0, 0}, {1,0,1, 2}, {0,0,0, 0}},   // B' R2
    {{1,0,1, 1}, {3,2,1, 0}, {0,0,0, 0}},   // B' R3
    {{0,3,1,11}, {0,0,0, 0}, {0,0,0, 0}}    // B' R4: wrap, MSB=bit0, LSB=bit12
  };

  double wr[16][16], wi[16][16];
  for (int a = 0; a < 16; ++a)
    for (int j = 0; j < 16; ++j) { wr[a][j] = (a == j) ? 1.0 : 0.0; wi[a][j] = 0.0; }

  for (int g = 0; g < NG[rid]; ++g) {
    const int bm = GT[rid][g][0], bl = GT[rid][g][1];
    const int off = GT[rid][g][2], ui = GT[rid][g][3];
    const cplx* U = Uws + ((layer * 2 + off) * 12 + ui) * 16;
    const int mask = (1 << bm) | (1 << bl);
    for (int j = 0; j < 16; ++j) {
      for (int b = 0; b < 16; ++b) {
        if (b & mask) continue;
        int idx[4] = {b, b | (1 << bl), b | (1 << bm), b | (1 << bm) | (1 << bl)};
        double xr[4], xim[4];
        for (int q = 0; q < 4; ++q) { xr[q] = wr[idx[q]][j]; xim[q] = wi[idx[q]][j]; }
        for (int r = 0; r < 4; ++r) {
          double sr = 0.0, si = 0.0;
          for (int q = 0; q < 4; ++q) {
            double urr = U[r * 4 + q].x, uii = U[r * 4 + q].y;
            sr += urr * xr[q] - uii * xim[q];
            si += urr * xim[q] + uii * xr[q];
          }
          wr[idx[r]][j] = sr; wi[idx[r]][j] = si;
        }
      }
    }
  }

  float* E = Eb + ro * 768;
  for (int a = 0; a < 16; ++a)
    for (int c = 0; c < 16; ++c) {
      double er = wr[a][c] - ((a == c) ? 1.0 : 0.0);
      double ei = wi[a][c];
      E[a * 16 + c]       = (float)er;
      E[256 + a * 16 + c] = (float)ei;
      E[512 + a * 16 + c] = (float)(-ei);
    }
}

// ---------------------------------------------------------------------------
// Pass A: tile = flat bits 0..11 (contiguous), 4096 amps, 256 thr (8 waves).
// ---------------------------------------------------------------------------
__global__ __launch_bounds__(256) void pass_a(const cplx* __restrict__ src,
                                              cplx* __restrict__ dst,
                                              const float* __restrict__ Eb,
                                              int layer) {
  extern __shared__ cplx tile[];
  const int tid = threadIdx.x;
  const unsigned blk = blockIdx.x;
  const unsigned long long gbase = (unsigned long long)blk << 12;

#pragma unroll
  for (int e = 0; e < 16; ++e) {
    int t = e * 256 + tid;
    async_g2l(lds_byte_addr(&tile[padidx(t)]), src + (gbase | (unsigned)t));
  }
  wait_async0();
  __syncthreads();

  const float* E = Eb + (layer * 11) * 768;
  const int lane = tid & 31, wave = tid >> 5;

  wmma_round<12, 0, 1, 2, 3>(tile, lane, wave, 2, E + 0 * 768);
  __syncthreads();
  wmma_round<12, 4, 5, 6, 7>(tile, lane, wave, 2, E + 1 * 768);
  __syncthreads();
  wmma_round<12, 8, 9, 10, 11>(tile, lane, wave, 2, E + 2 * 768);
  __syncthreads();
  wmma_round<12, 3, 4, 7, 8>(tile, lane, wave, 2, E + 3 * 768);
  __syncthreads();

#pragma unroll
  for (int e = 0; e < 16; ++e) {
    int t = e * 256 + tid;
    async_l2g(dst + (gbase | (unsigned)t), lds_byte_addr(&tile[padidx(t)]));
  }
  wait_async0();
}

// ---------------------------------------------------------------------------
// Pass B (variant 0): tile = bits {0..5} U {11..17}
// Pass B' (variant 1): tile = bits {0..5} U {17..23} (+ wrap gate round)
// 8192 amps, 512 thr (16 waves).
// ---------------------------------------------------------------------------
__device__ __forceinline__ unsigned long long gidx_b0(unsigned t, unsigned blk) {
  return (unsigned long long)((t & 63u) | ((blk & 31u) << 6) |
                              (((t >> 6) & 127u) << 11) | ((blk >> 5) << 18));
}
__device__ __forceinline__ unsigned long long gidx_b1(unsigned t, unsigned blk) {
  return (unsigned long long)((t & 63u) | (blk << 6) | (((t >> 6) & 127u) << 17));
}

__global__ __launch_bounds__(512) void pass_b(const cplx* __restrict__ src,
                                              cplx* __restrict__ dst,
                                              const float* __restrict__ Eb,
                                              int layer, int variant) {
  extern __shared__ cplx tile[];
  const int tid = threadIdx.x;
  const unsigned blk = blockIdx.x;

#pragma unroll
  for (int e = 0; e < 16; ++e) {
    unsigned t = (unsigned)(e * 512 + tid);
    unsigned long long g = variant ? gidx_b1(t, blk) : gidx_b0(t, blk);
    async_g2l(lds_byte_addr(&tile[padidx((int)t)]), src + g);
  }
  wait_async0();
  __syncthreads();

  const float* E = Eb + (layer * 11 + (variant ? 7 : 4)) * 768;
  const int lane = tid & 31, wave = tid >> 5;

  wmma_round<13, 7, 8, 9, 10>(tile, lane, wave, 2, E + 0 * 768);
  __syncthreads();
  wmma_round<13, 6, 7, 11, 12>(tile, lane, wave, 2, E + 1 * 768);
  __syncthreads();
  wmma_round<13, 8, 9, 10, 11>(tile, lane, wave, 2, E + 2 * 768);
  __syncthreads();
  if (variant) {
    wmma_round<13, 0, 1, 2, 12>(tile, lane, wave, 2, E + 3 * 768);
    __syncthreads();
  }

#pragma unroll
  for (int e = 0; e < 16; ++e) {
    unsigned t = (unsigned)(e * 512 + tid);
    unsigned long long g = variant ? gidx_b1(t, blk) : gidx_b0(t, blk);
    async_l2g(dst + g, lds_byte_addr(&tile[padidx((int)t)]));
  }
  wait_async0();
}

// ---------------------------------------------------------------------------
extern "C" void kernel_launch(void* const* d_in, const int* in_sizes, int n_in,
                              void* d_out, int out_size, void* d_ws, size_t ws_size,
                              hipStream_t stream) {
  const cplx* state = (const cplx*)d_in[0];
  const cplx* H1 = (const cplx*)d_in[1];
  const cplx* H2 = (const cplx*)d_in[2];
  cplx* out = (cplx*)d_out;
  cplx* Uws = (cplx*)d_ws;                                   // 48*16 cplx = 12 KB
  float* Eb = (float*)((char*)d_ws + 48 * 16 * sizeof(cplx)); // 22*768 f32 = 66 KB

  (void)in_sizes; (void)n_in; (void)out_size; (void)ws_size;

  expm_kernel<<<1, 64, 0, stream>>>(H1, H2, Uws);
  roundop_kernel<<<1, 32, 0, stream>>>(Uws, Eb);

  const size_t LDS_A = (size_t)(4096 + 256) * sizeof(cplx);   // 69632 B
  const size_t LDS_B = (size_t)(8192 + 512) * sizeof(cplx);   // 139264 B

  for (int layer = 0; layer < 2; ++layer) {
    const cplx* s = (layer == 0) ? state : out;
    pass_a<<<4096, 256, LDS_A, stream>>>(s, out, Eb, layer);
    pass_b<<<2048, 512, LDS_B, stream>>>(out, out, Eb, layer, 0);
    pass_b<<<2048, 512, LDS_B, stream>>>(out, out, Eb, layer, 1);
  }
}